// GraphAttention_34737695490297
// MI455X (gfx1250) — compile-verified
//
#include <hip/hip_runtime.h>
#include <math.h>

typedef float v2f __attribute__((ext_vector_type(2)));
typedef float v8f __attribute__((ext_vector_type(8)));

#define N_NODES   8192
#define F_IN      512
#define F_OUT     128
#define NEG_SLOPE 0.2f
#define MASK_VAL  -9e16f

__device__ __forceinline__ float leaky(float x) {
    return x > 0.f ? x : NEG_SLOPE * x;
}

// ---------------------------------------------------------------------------
// Kernel 1: h = x @ w   (f32 WMMA 16x16x4, exact f32 precision)
// Also writes hT = h^T so the big GEMM's B fragments are contiguous b64 loads.
// One wave per 16x16 tile; 8 waves/block share the 16-row stripe of x.
// ---------------------------------------------------------------------------
__global__ __launch_bounds__(256) void gat_gemm_h(const float* __restrict__ x,
                                                  const float* __restrict__ w,
                                                  float* __restrict__ h,
                                                  float* __restrict__ hT) {
    const int r0   = blockIdx.x * 16;
    const int wave = threadIdx.x >> 5;
    const int lane = threadIdx.x & 31;
    const int l15  = lane & 15;         // A row (M) and B column (N)
    const int half = lane >> 4;         // K-pair selector
    const int n0   = wave * 16;

    v8f acc = {0.f, 0.f, 0.f, 0.f, 0.f, 0.f, 0.f, 0.f};
    const float* xrow = x + (size_t)(r0 + l15) * F_IN;

#pragma unroll 4
    for (int k = 0; k < F_IN; k += 4) {
        const int k0 = k + 2 * half;
        v2f a = *(const v2f*)(xrow + k0);                  // A[m][k0], A[m][k0+1]
        v2f b;
        b.x = w[(size_t)k0 * F_OUT + n0 + l15];            // B[k0][n]
        b.y = w[(size_t)(k0 + 1) * F_OUT + n0 + l15];      // B[k0+1][n]
        acc = __builtin_amdgcn_wmma_f32_16x16x4_f32(false, a, false, b,
                                                    (short)0, acc, false, false);
    }
#pragma unroll
    for (int v = 0; v < 8; ++v) {
        const int rr = r0 + v + 8 * half;
        const int cc = n0 + l15;
        h [(size_t)rr * F_OUT   + cc] = acc[v];
        hT[(size_t)cc * N_NODES + rr] = acc[v];
    }
}

// ---------------------------------------------------------------------------
// Kernel 2: s = h @ a[:128], t = h @ a[128:]  (one wave per row, wave32 reduce)
// ---------------------------------------------------------------------------
__global__ __launch_bounds__(256) void gat_scores(const float* __restrict__ h,
                                                  const float* __restrict__ a,
                                                  float* __restrict__ s,
                                                  float* __restrict__ t) {
    const int row  = (int)((blockIdx.x * blockDim.x + threadIdx.x) >> 5);
    const int lane = threadIdx.x & 31;
    const float* hr = h + (size_t)row * F_OUT;

    float ss = 0.f, tt = 0.f;
#pragma unroll
    for (int c = lane; c < F_OUT; c += 32) {
        float hv = hr[c];
        ss += hv * a[c];
        tt += hv * a[F_OUT + c];
    }
#pragma unroll
    for (int off = 16; off > 0; off >>= 1) {
        ss += __shfl_xor(ss, off, 32);
        tt += __shfl_xor(tt, off, 32);
    }
    if (lane == 0) { s[row] = ss; t[row] = tt; }
}

// ---------------------------------------------------------------------------
// Kernel 3: masked leaky-relu logits + row softmax -> attention.
// One block per row; three vectorized passes over the (L2-hot) adj row:
//   A) max (no exp), B) sum of exp (one __expf/elem), C) write probabilities.
// Masked entries carry MASK_VAL (matches reference exactly, incl. fully
// masked rows -> uniform). __expf(MASK_VAL - M) underflows to exactly 0.
// ---------------------------------------------------------------------------
__global__ __launch_bounds__(256) void gat_softmax(const int* __restrict__ adj,
                                                   const float* __restrict__ s,
                                                   const float* __restrict__ t,
                                                   float* __restrict__ attn) {
    const int row = blockIdx.x;
    const int tid = threadIdx.x;
    const float si = s[row];
    const int4*   arow4 = (const int4*)(adj + (size_t)row * N_NODES);
    const float4* t4    = (const float4*)t;
    __shared__ float red[256];

    // ---- Pass A: row maximum ----
    float m = -INFINITY;
    for (int j = tid; j < N_NODES / 4; j += 256) {
        int4   av = arow4[j];
        float4 tv = t4[j];
        float e0 = av.x > 0 ? leaky(si + tv.x) : MASK_VAL;
        float e1 = av.y > 0 ? leaky(si + tv.y) : MASK_VAL;
        float e2 = av.z > 0 ? leaky(si + tv.z) : MASK_VAL;
        float e3 = av.w > 0 ? leaky(si + tv.w) : MASK_VAL;
        m = fmaxf(m, fmaxf(fmaxf(e0, e1), fmaxf(e2, e3)));
    }
    red[tid] = m;
    __syncthreads();
    for (int st = 128; st > 0; st >>= 1) {
        if (tid < st) red[tid] = fmaxf(red[tid], red[tid + st]);
        __syncthreads();
    }
    const float M = red[0];
    __syncthreads();

    // ---- Pass B: sum of exp(e - M) ----
    float z = 0.f;
    for (int j = tid; j < N_NODES / 4; j += 256) {
        int4   av = arow4[j];
        float4 tv = t4[j];
        float e0 = av.x > 0 ? leaky(si + tv.x) : MASK_VAL;
        float e1 = av.y > 0 ? leaky(si + tv.y) : MASK_VAL;
        float e2 = av.z > 0 ? leaky(si + tv.z) : MASK_VAL;
        float e3 = av.w > 0 ? leaky(si + tv.w) : MASK_VAL;
        z += __expf(e0 - M) + __expf(e1 - M) + __expf(e2 - M) + __expf(e3 - M);
    }
    red[tid] = z;
    __syncthreads();
    for (int st = 128; st > 0; st >>= 1) {
        if (tid < st) red[tid] += red[tid + st];
        __syncthreads();
    }
    const float invZ = 1.f / red[0];

    // ---- Pass C: write normalized attention ----
    float4* orow4 = (float4*)(attn + (size_t)row * N_NODES);
    for (int j = tid; j < N_NODES / 4; j += 256) {
        int4   av = arow4[j];
        float4 tv = t4[j];
        float e0 = av.x > 0 ? leaky(si + tv.x) : MASK_VAL;
        float e1 = av.y > 0 ? leaky(si + tv.y) : MASK_VAL;
        float e2 = av.z > 0 ? leaky(si + tv.z) : MASK_VAL;
        float e3 = av.w > 0 ? leaky(si + tv.w) : MASK_VAL;
        float4 o;
        o.x = __expf(e0 - M) * invZ;
        o.y = __expf(e1 - M) * invZ;
        o.z = __expf(e2 - M) * invZ;
        o.w = __expf(e3 - M) * invZ;
        orow4[j] = o;
    }
}

// ---------------------------------------------------------------------------
// Kernel 4: h_prime = attention @ h   (f32 WMMA 16x16x4, K = 8192)
// B fragments come from hT so both A and B are single b64 loads per lane.
// ---------------------------------------------------------------------------
__global__ __launch_bounds__(256) void gat_gemm_out(const float* __restrict__ attn,
                                                    const float* __restrict__ hT,
                                                    float* __restrict__ hp) {
    const int r0   = blockIdx.x * 16;
    const int wave = threadIdx.x >> 5;
    const int lane = threadIdx.x & 31;
    const int l15  = lane & 15;
    const int half = lane >> 4;
    const int n0   = wave * 16;

    v8f acc = {0.f, 0.f, 0.f, 0.f, 0.f, 0.f, 0.f, 0.f};
    const float* arow = attn + (size_t)(r0 + l15) * N_NODES;   // A row (M = l15)
    const float* bcol = hT + (size_t)(n0 + l15) * N_NODES;     // B col (N = l15)

#pragma unroll 4
    for (int k = 0; k < N_NODES; k += 4) {
        const int k0 = k + 2 * half;
        v2f a = *(const v2f*)(arow + k0);   // A[m][k0], A[m][k0+1]
        v2f b = *(const v2f*)(bcol + k0);   // B[k0][n], B[k0+1][n]
        acc = __builtin_amdgcn_wmma_f32_16x16x4_f32(false, a, false, b,
                                                    (short)0, acc, false, false);
    }
#pragma unroll
    for (int v = 0; v < 8; ++v) {
        hp[(size_t)(r0 + v + 8 * half) * F_OUT + n0 + l15] = acc[v];
    }
}

// ---------------------------------------------------------------------------
extern "C" void kernel_launch(void* const* d_in, const int* in_sizes, int n_in,
                              void* d_out, int out_size, void* d_ws, size_t ws_size,
                              hipStream_t stream) {
    const float* x   = (const float*)d_in[0];   // [8192, 512]
    const int*   adj = (const int*)  d_in[1];   // [8192, 8192]
    const float* w   = (const float*)d_in[2];   // [512, 128]
    const float* a   = (const float*)d_in[3];   // [256, 1]

    float* attn = (float*)d_out;                               // [8192, 8192]
    float* hp   = (float*)d_out + (size_t)N_NODES * N_NODES;   // [8192, 128]

    float* h  = (float*)d_ws;                                  // [8192, 128]
    float* hT = h  + (size_t)N_NODES * F_OUT;                  // [128, 8192]
    float* s  = hT + (size_t)N_NODES * F_OUT;                  // [8192]
    float* t  = s + N_NODES;                                   // [8192]

    gat_gemm_h  <<<N_NODES / 16, 256, 0, stream>>>(x, w, h, hT);
    gat_scores  <<<N_NODES / 8,  256, 0, stream>>>(h, a, s, t);
    gat_softmax <<<N_NODES,      256, 0, stream>>>(adj, s, t, attn);
    gat_gemm_out<<<N_NODES / 16, 256, 0, stream>>>(attn, hT, hp);
}